// HMGConvolution_10711648436917
// MI455X (gfx1250) — compile-verified
//
#include <hip/hip_runtime.h>

typedef __attribute__((ext_vector_type(2))) float v2f;
typedef __attribute__((ext_vector_type(8))) float v8f;

#define N_NODES   100000
#define N_SUPPORTS 3
#define N_EDGES   500000
#define DIM       128
#define M_TILES   (N_NODES / 16)   // 6250, N_NODES divisible by 16

// ---------------------------------------------------------------------------
// Zero the [N,128] scratch accumulator (float4 stores).
// ---------------------------------------------------------------------------
__global__ void hmg_zero_kernel(float4* __restrict__ p, int n4) {
    int i = blockIdx.x * blockDim.x + threadIdx.x;
    if (i < n4) p[i] = make_float4(0.f, 0.f, 0.f, 0.f);
}

// ---------------------------------------------------------------------------
// SpMM:  tmp[dst] += val * x[src]   (one wave32 per edge; lane owns 4 floats)
// x rows are 512B; lane loads float4 at lane*16B -> fully coalesced gather.
// fp32 atomics land in L2 (x and tmp both resident in 192MB L2).
// ---------------------------------------------------------------------------
__global__ void hmg_spmm_kernel(const float* __restrict__ x,
                                const int*   __restrict__ src,
                                const int*   __restrict__ dst,
                                const float* __restrict__ val,
                                float* __restrict__ tmp) {
    int edge = (blockIdx.x * blockDim.x + threadIdx.x) >> 5;  // wave-uniform
    int lane = threadIdx.x & 31;
    if (edge >= N_EDGES) return;

    int   s = src[edge];
    int   d = dst[edge];
    float v = val[edge];

    const float4* xs = (const float4*)(x + (size_t)s * DIM);
    float4 m = xs[lane];

    float* t = tmp + (size_t)d * DIM + lane * 4;
    unsafeAtomicAdd(t + 0, m.x * v);
    unsafeAtomicAdd(t + 1, m.y * v);
    unsafeAtomicAdd(t + 2, m.z * v);
    unsafeAtomicAdd(t + 3, m.w * v);
}

// ---------------------------------------------------------------------------
// GEMM:  out[m,:] (+)= tmp[m,:] @ W   via V_WMMA_F32_16X16X4_F32.
// One wave computes a 16(M) x 128(N) strip: 8 accumulator tiles, K in steps
// of 4 (32 iterations, 8 wmma each). A-frag loaded once per k-step.
// accumulate==0 -> overwrite (support 0); relu applied on last support.
// ---------------------------------------------------------------------------
__global__ void hmg_gemm_kernel(const float* __restrict__ A,   // tmp [N,128]
                                const float* __restrict__ B,   // W_s [128,128]
                                float* __restrict__ out,       // [N,128]
                                int accumulate, int relu) {
    int wave = blockIdx.x * (blockDim.x >> 5) + (threadIdx.x >> 5); // uniform
    if (wave >= M_TILES) return;                                   // uniform exit

    int lane = threadIdx.x & 31;
    int half = lane >> 4;      // 0: lanes 0-15, 1: lanes 16-31
    int lr   = lane & 15;
    int m0   = wave * 16;

    v8f acc[8] = {};

    const float* Arow = A + (size_t)(m0 + lr) * DIM + 2 * half;
    const float* Brow = B + (size_t)(2 * half) * DIM + lr;

    for (int k0 = 0; k0 < DIM; k0 += 4) {
        // A fragment: lane holds A[m0+lr][k0 + 2*half + {0,1}]
        float2 af = *(const float2*)(Arow + k0);
        v2f a; a.x = af.x; a.y = af.y;

        const float* Bk = Brow + (size_t)k0 * DIM;
#pragma unroll
        for (int t = 0; t < 8; t++) {
            int n0 = t * 16;
            v2f b;
            b.x = Bk[n0];          // B[k0 + 2*half    ][n0 + lr]
            b.y = Bk[n0 + DIM];    // B[k0 + 2*half + 1][n0 + lr]
            acc[t] = __builtin_amdgcn_wmma_f32_16x16x4_f32(
                false, a, false, b, (short)0, acc[t], false, false);
        }
    }

#pragma unroll
    for (int t = 0; t < 8; t++) {
        int n0 = t * 16;
#pragma unroll
        for (int r = 0; r < 8; r++) {
            int row = m0 + r + 8 * half;
            size_t idx = (size_t)row * DIM + n0 + lr;
            float res = acc[t][r];
            if (accumulate) res += out[idx];
            if (relu)       res = fmaxf(res, 0.f);
            out[idx] = res;
        }
    }
}

// ---------------------------------------------------------------------------
// Launch: per support s: zero tmp -> SpMM(A_s @ x) -> WMMA GEMM (@ W_s).
// out = relu fused into the final GEMM pass.
// ---------------------------------------------------------------------------
extern "C" void kernel_launch(void* const* d_in, const int* in_sizes, int n_in,
                              void* d_out, int out_size, void* d_ws, size_t ws_size,
                              hipStream_t stream) {
    const float* x    = (const float*)d_in[0];
    const float* W    = (const float*)d_in[1];
    const int*   esrc = (const int*)  d_in[2];
    const int*   edst = (const int*)  d_in[3];
    const float* eval = (const float*)d_in[4];
    float* out = (float*)d_out;
    float* tmp = (float*)d_ws;   // needs N_NODES*DIM*4 = 51.2 MB

    const int n4 = N_NODES * DIM / 4;
    const int zero_blocks = (n4 + 255) / 256;
    const int spmm_blocks = (N_EDGES + 7) / 8;      // 8 waves (edges) per block
    const int gemm_blocks = (M_TILES + 7) / 8;      // 8 waves per block

    for (int s = 0; s < N_SUPPORTS; s++) {
        hmg_zero_kernel<<<zero_blocks, 256, 0, stream>>>((float4*)tmp, n4);
        hmg_spmm_kernel<<<spmm_blocks, 256, 0, stream>>>(
            x, esrc + (size_t)s * N_EDGES, edst + (size_t)s * N_EDGES,
            eval + (size_t)s * N_EDGES, tmp);
        hmg_gemm_kernel<<<gemm_blocks, 256, 0, stream>>>(
            tmp, W + (size_t)s * DIM * DIM, out,
            /*accumulate=*/(s > 0), /*relu=*/(s == N_SUPPORTS - 1));
    }
}